// NormalBasedSDFLoss_24137716204097
// MI455X (gfx1250) — compile-verified
//
#include <hip/hip_runtime.h>
#include <hip/hip_bf16.h>

typedef __attribute__((ext_vector_type(2))) float v2f;
typedef __attribute__((ext_vector_type(8))) float v8f;

#define SDF_W 7000.0f
#define EIK_W 600.0f
#define ORI_W 500.0f
#define NEAR_ORI_W 10.0f
#define GRADN_W 200.0f

// ---------------------------------------------------------------------------
// Kernel A: elementwise reductions (sdf, eikonal, grad-normal), per-wave
// partials written to fixed slots (deterministic, no atomics).
// ---------------------------------------------------------------------------
__global__ void reduce_elem_kernel(const float* __restrict__ mpred,   // [N,1]
                                   const float* __restrict__ mgrad,   // [N,3]
                                   const float* __restrict__ snrm,    // [N,3]
                                   float* __restrict__ psdf,
                                   float* __restrict__ peik,
                                   float* __restrict__ pgrad,
                                   int N) {
  const int tid    = blockIdx.x * blockDim.x + threadIdx.x;
  const int stride = gridDim.x * blockDim.x;
  float s = 0.0f, e = 0.0f, g = 0.0f;
  for (int i = tid; i < N; i += stride) {
    float p = mpred[i];
    s += p * p;
    float gx = mgrad[3 * i + 0], gy = mgrad[3 * i + 1], gz = mgrad[3 * i + 2];
    float nrm = sqrtf(gx * gx + gy * gy + gz * gz);
    float d = nrm - 1.0f;
    e += d * d;
    float nx = snrm[3 * i + 0], ny = snrm[3 * i + 1], nz = snrm[3 * i + 2];
    float dx = gx - nx, dy = gy - ny, dz = gz - nz;
    g += dx * dx + dy * dy + dz * dz;
  }
  #pragma unroll
  for (int off = 16; off >= 1; off >>= 1) {
    s += __shfl_xor(s, off, 32);
    e += __shfl_xor(e, off, 32);
    g += __shfl_xor(g, off, 32);
  }
  const int wave = tid >> 5;
  if ((threadIdx.x & 31) == 0) {
    psdf[wave]  = s;
    peik[wave]  = e;
    pgrad[wave] = g;
  }
}

// ---------------------------------------------------------------------------
// Kernel B: WMMA-based brute-force 2-NN + orientation loss.
// One wave handles 16 queries. Score = q.s - 0.5|s|^2 via V_WMMA_F32_16X16X4:
//   A (16x4): [qx qy qz 1]   lanes 0-15 -> K=0,1 ; lanes 16-31 -> K=2,3
//   B (4x16): [sx sy sz -.5|s|^2], mirrored layout, N = lane%16
//   C (16x16): VGPR v of lane L holds (M = v + 8*(L>=16), N = L%16)
// Per-lane running top-2, all updates BRANCHLESS (v_cndmask, no EXEC writes),
// XOR-butterfly merge within each 16-lane half, lanes 0/16 finalize.
// ---------------------------------------------------------------------------
__global__ void ori_knn_kernel(const float* __restrict__ query,  // [N,3]
                               const float* __restrict__ pred,   // [N,1]
                               const float* __restrict__ spts,   // [N,3]
                               const float* __restrict__ snrm,   // [N,3]
                               float* __restrict__ partial,      // [N/16]
                               int N) {
  const int lane  = threadIdx.x & 31;
  const int wave  = (blockIdx.x * blockDim.x + threadIdx.x) >> 5;
  const int qbase = wave * 16;
  if (qbase >= N) return;  // wave-uniform: EXEC stays all-ones for WMMA

  const int  col = lane & 15;
  const bool hi  = lane >= 16;

  // Load A tile (constant for the whole scan)
  const int m0 = qbase + col;
  const float qx = query[3 * m0 + 0];
  const float qy = query[3 * m0 + 1];
  const float qz = query[3 * m0 + 2];
  v2f a;
  a.x = hi ? qz : qx;     // K=2 : K=0
  a.y = hi ? 1.0f : qy;   // K=3 : K=1

  float best[8], second[8];
  int   bidx[8], sidx[8];
  #pragma unroll
  for (int v = 0; v < 8; ++v) {
    best[v]   = -__builtin_inff();
    second[v] = -__builtin_inff();
    bidx[v]   = 0;
    sidx[v]   = 0;
  }

  const float* sp = spts + 3 * col;    // this lane's column pointer
  #pragma unroll 2
  for (int tile = 0; tile < N; tile += 16) {
    const int j = tile + col;
    const float sx = sp[0];
    const float sy = sp[1];
    const float sz = sp[2];
    __builtin_prefetch(sp + 48, 0, 0);  // speculative: OOB is silently dropped
    sp += 48;                           // 16 points * 3 floats
    const float hb = -0.5f * (sx * sx + sy * sy + sz * sz);
    v2f b;
    b.x = hi ? sz : sx;   // K=2 : K=0
    b.y = hi ? hb : sy;   // K=3 : K=1

    v8f c = {};
    c = __builtin_amdgcn_wmma_f32_16x16x4_f32(
        /*neg_a=*/false, a, /*neg_b=*/false, b,
        /*c_mod=*/(short)0, c, /*reuse_a=*/false, /*reuse_b=*/false);

    #pragma unroll
    for (int v = 0; v < 8; ++v) {
      const float sc = c[v];
      const bool gb = sc > best[v];     // beats current best
      const bool gs = sc > second[v];   // beats current second
      const float nsec = gb ? best[v] : (gs ? sc : second[v]);
      const int   nsix = gb ? bidx[v] : (gs ? j  : sidx[v]);
      const float nbst = gb ? sc : best[v];
      const int   nbix = gb ? j  : bidx[v];
      second[v] = nsec; sidx[v] = nsix;
      best[v]   = nbst; bidx[v] = nbix;
    }
  }

  // Merge the 16 per-lane top-2 lists within each half (offsets keep bit4).
  #pragma unroll
  for (int off = 1; off <= 8; off <<= 1) {
    #pragma unroll
    for (int v = 0; v < 8; ++v) {
      const float b1 = __shfl_xor(best[v],   off, 32);
      const int   j1 = __shfl_xor(bidx[v],   off, 32);
      const float b2 = __shfl_xor(second[v], off, 32);
      const int   j2 = __shfl_xor(sidx[v],   off, 32);
      // merged top-2 of {best,second} and {b1,b2} (each pair sorted desc)
      const bool t  = b1 > best[v];
      const float nbst = t ? b1 : best[v];
      const int   nbix = t ? j1 : bidx[v];
      const float m1 = t ? best[v] : b1;     // loser of the firsts
      const int   i1 = t ? bidx[v] : j1;
      const float m2 = t ? b2 : second[v];   // winner's second
      const int   i2 = t ? j2 : sidx[v];
      const bool u  = m2 > m1;
      second[v] = u ? m2 : m1;
      sidx[v]   = u ? i2 : i1;
      best[v]   = nbst;
      bidx[v]   = nbix;
    }
  }

  // Lanes 0 (queries 0-7) and 16 (queries 8-15) finalize.
  float lsum = 0.0f;
  if (col == 0) {
    #pragma unroll
    for (int v = 0; v < 8; ++v) {
      const int mq = qbase + v + (hi ? 8 : 0);
      const float ax = query[3 * mq + 0];
      const float ay = query[3 * mq + 1];
      const float az = query[3 * mq + 2];
      float dsum = 0.0f;
      const int idx2[2] = {bidx[v], sidx[v]};
      #pragma unroll
      for (int kk = 0; kk < 2; ++kk) {
        const int ii = idx2[kk];
        const float px = spts[3 * ii + 0], py = spts[3 * ii + 1], pz = spts[3 * ii + 2];
        const float nx = snrm[3 * ii + 0], ny = snrm[3 * ii + 1], nz = snrm[3 * ii + 2];
        dsum += (ax - px) * nx + (ay - py) * ny + (az - pz) * nz;
      }
      const float mean = 0.5f * dsum;
      const float sgn  = (mean > 0.0f) ? 1.0f : ((mean < 0.0f) ? -1.0f : 0.0f);
      const float t    = -pred[mq] * sgn;
      lsum += (t > 0.0f) ? t : 0.0f;
    }
  }
  #pragma unroll
  for (int off = 16; off >= 1; off >>= 1) lsum += __shfl_xor(lsum, off, 32);
  if (lane == 0) partial[wave] = lsum;
}

// ---------------------------------------------------------------------------
// Kernel C: deterministic serial finalize of all partials -> 6 outputs.
// ---------------------------------------------------------------------------
__global__ void finalize_kernel(const float* __restrict__ psdf,
                                const float* __restrict__ peik,
                                const float* __restrict__ pgrad,
                                const float* __restrict__ pori,
                                const float* __restrict__ pnori,
                                int rw, int ow, int N,
                                float* __restrict__ out) {
  float ssum = 0.0f, esum = 0.0f, gsum = 0.0f, osum = 0.0f, nsum = 0.0f;
  for (int i = 0; i < rw; ++i) { ssum += psdf[i]; esum += peik[i]; gsum += pgrad[i]; }
  for (int i = 0; i < ow; ++i) { osum += pori[i]; nsum += pnori[i]; }
  const float invN = 1.0f / (float)N;
  const float sdf   = ssum * invN;
  const float eik   = esum * invN;
  const float gradn = gsum / (3.0f * (float)N);
  const float ori   = osum * invN;
  const float nori  = nsum * invN;
  out[0] = SDF_W * sdf + EIK_W * eik + ORI_W * ori + NEAR_ORI_W * nori + GRADN_W * gradn;
  out[1] = sdf;
  out[2] = eik;
  out[3] = ori;
  out[4] = nori;
  out[5] = gradn;
}

extern "C" void kernel_launch(void* const* d_in, const int* in_sizes, int n_in,
                              void* d_out, int out_size, void* d_ws, size_t ws_size,
                              hipStream_t stream) {
  const float* manifold_pred    = (const float*)d_in[0];
  const float* manifold_grad    = (const float*)d_in[1];
  const float* nonmanifold_pred = (const float*)d_in[2];
  const float* near_points_pred = (const float*)d_in[3];
  const float* surface_points   = (const float*)d_in[4];
  const float* surface_normals  = (const float*)d_in[5];
  const float* off_surface      = (const float*)d_in[6];
  const float* near_points      = (const float*)d_in[7];
  float* out = (float*)d_out;

  const int N = in_sizes[0];          // 16384 rows
  const int RBLK = 64, RTHD = 256;
  const int RW = (RBLK * RTHD) / 32;  // 512 reduce waves
  const int OW = N / 16;              // 1024 kNN waves per sign pass

  float* ws    = (float*)d_ws;
  float* psdf  = ws;
  float* peik  = ws + RW;
  float* pgrad = ws + 2 * RW;
  float* pori  = ws + 3 * RW;
  float* pnori = pori + OW;

  reduce_elem_kernel<<<RBLK, RTHD, 0, stream>>>(
      manifold_pred, manifold_grad, surface_normals, psdf, peik, pgrad, N);

  const int oriBlocks = (OW * 32 + 255) / 256;
  ori_knn_kernel<<<oriBlocks, 256, 0, stream>>>(
      off_surface, nonmanifold_pred, surface_points, surface_normals, pori, N);
  ori_knn_kernel<<<oriBlocks, 256, 0, stream>>>(
      near_points, near_points_pred, surface_points, surface_normals, pnori, N);

  finalize_kernel<<<1, 1, 0, stream>>>(psdf, peik, pgrad, pori, pnori, RW, OW, N, out);
}